// RSKH_89644557402450
// MI455X (gfx1250) — compile-verified
//
#include <hip/hip_runtime.h>
#include <math.h>

typedef __attribute__((ext_vector_type(2))) float v2f;
typedef __attribute__((ext_vector_type(8))) float v8f;

#define B_DIM   16
#define S_DIM   8192
#define K_DIM   64          // knots per row (each knot = float4)
#define STILE   128         // s-rows per block in stage 1
#define NTILES  (S_DIM / STILE)   // 64 s-tiles per batch

// ---------------------------------------------------------------------------
// Stage 1: streaming eta computation + per-(b,stile,k) partial sums.
// Grid: B_DIM*NTILES blocks of 256 threads (8 wave32).
// Thread t: knot k = t&63, row phase = t>>6. 64 consecutive threads read one
// contiguous 1KiB row segment as float4 -> global_load_b128, fully coalesced.
// ---------------------------------------------------------------------------
__global__ __launch_bounds__(256) void eta_partial_kernel(
    const float4* __restrict__ x, float* __restrict__ partial)
{
    __shared__ float red[256];
    const int t     = threadIdx.x;
    const int k     = t & 63;
    const int srow  = t >> 6;          // 0..3
    const int b     = blockIdx.x >> 6; // /NTILES
    const int stile = blockIdx.x & 63;

    const float4* row0 = x + (size_t)(b * S_DIM + stile * STILE) * K_DIM;
    const float inv_pi = 0.31830988618379067154f;

    float acc = 0.0f;
    #pragma unroll 4
    for (int r = 0; r < STILE / 4; ++r) {
        const int s_local = r * 4 + srow;
        float4 q = row0[(size_t)s_local * K_DIM + k];
        float a = q.x, bb = q.y, c = q.z, d = q.w;

        // U = [[a+ib, c+id], [-c+id, a-ib]],  S = U^H U (transcribed op-by-op)
        float s00  = (a*a + bb*bb) + (c*c + d*d);      // imag(s00) == 0 exactly
        float s11  = (c*c + d*d) + (a*a + bb*bb);      // imag(s11) == 0 exactly
        float s01r = (a*c + bb*d) + (-(c*a) - d*bb);
        float s01i = (a*d - bb*c) + (c*bb - d*a);
        float s10r = (c*a + d*bb) + (-(a*c) - bb*d);
        float s10i = (c*bb - d*a) + (a*d - bb*c);

        float detr = s00 * s11 - (s01r * s10r - s01i * s10i);
        float deti = -(s01r * s10i + s01i * s10r);

        acc += atan2f(deti, detr) * inv_pi;
    }

    red[t] = acc;
    __syncthreads();
    if (t < 64) {   // fold the 4 row phases per knot -> one partial per (b,stile,k)
        float v = red[t] + red[t + 64] + red[t + 128] + red[t + 192];
        partial[(size_t)blockIdx.x * 64 + t] = v;   // [(b*NTILES+stile)*64 + k]
    }
}

// ---------------------------------------------------------------------------
// Stage 2: deterministic 64-way reduction per output via fp32 WMMA.
// For each batch b, partial is a 64(stile) x 64(k) matrix; column sums are
// computed as D += ones(16x4) * B(4x16) accumulated over 16 chunks.
// One wave32 per (b, 16-knot group): grid = 64 blocks of 32 threads.
// B-matrix 4x16 f32 layout: lanes 0-15 = N (knot), VGPR0 holds K rows {0,2}
// (lo/hi half-wave), VGPR1 holds K rows {1,3} — mirror of the documented
// 16x4 A layout. A = all ones, so D's row 0 = column sums over all stiles.
// ---------------------------------------------------------------------------
__global__ __launch_bounds__(32) void eta_reduce_wmma(
    const float* __restrict__ partial, float* __restrict__ out)
{
    const int lane = threadIdx.x;
    const int b    = blockIdx.x >> 2;
    const int kg   = blockIdx.x & 3;
    const int n    = lane & 15;
    const int hi   = lane >> 4;          // half-wave selects K rows {0,1} vs {2,3}
    const int k    = kg * 16 + n;

    const float* base = partial + (size_t)b * (NTILES * 64);

    v2f A; A[0] = 1.0f; A[1] = 1.0f;     // 16x4 matrix of ones
    v8f acc = {};

    #pragma unroll
    for (int c = 0; c < 16; ++c) {
        const int r0 = c * 4 + 2 * hi;   // stile row for K slot in VGPR0
        v2f Bm;
        Bm[0] = base[(size_t)r0 * 64 + k];        // K row 2*hi
        Bm[1] = base[(size_t)(r0 + 1) * 64 + k];  // K row 2*hi+1
        acc = __builtin_amdgcn_wmma_f32_16x16x4_f32(
            /*neg_a=*/false, A, /*neg_b=*/false, Bm,
            /*c_mod=*/(short)0, acc, /*reuse_a=*/false, /*reuse_b=*/false);
    }

    // D row 0 (lanes 0-15, VGPR0) holds the per-knot sums over all 64 stiles.
    if (lane < 16) {
        out[b * 64 + k] = acc[0] * (1.0f / (float)S_DIM);
    }
}

// ---------------------------------------------------------------------------
extern "C" void kernel_launch(void* const* d_in, const int* in_sizes, int n_in,
                              void* d_out, int out_size, void* d_ws, size_t ws_size,
                              hipStream_t stream) {
    (void)in_sizes; (void)n_in; (void)out_size; (void)ws_size;
    const float4* x  = (const float4*)d_in[0];
    float* out       = (float*)d_out;
    float* partials  = (float*)d_ws;   // B_DIM*NTILES*64 floats = 256 KiB

    eta_partial_kernel<<<dim3(B_DIM * NTILES), dim3(256), 0, stream>>>(x, partials);
    eta_reduce_wmma  <<<dim3(B_DIM * 4),       dim3(32),  0, stream>>>(partials, out);
}